// negative_sample_44504451121560
// MI455X (gfx1250) — compile-verified
//
#include <hip/hip_runtime.h>
#include <stdint.h>

#define N_NODES   4096
#define E_EDGES   16384
#define HALF_E    (E_EDGES / 2)
#define D_FEAT    256
#define MAX_ROUNDS 4

typedef __attribute__((ext_vector_type(2))) float v2f;
typedef __attribute__((ext_vector_type(8))) float v8f;

// ---------------- JAX-compatible threefry2x32 ----------------
__device__ __forceinline__ uint32_t rotl32(uint32_t x, uint32_t r) {
  return (x << r) | (x >> (32u - r));
}

__device__ __forceinline__ void threefry2x32(uint32_t k0, uint32_t k1,
                                             uint32_t x0, uint32_t x1,
                                             uint32_t& o0, uint32_t& o1) {
  const uint32_t ks2 = k0 ^ k1 ^ 0x1BD11BDAu;
  uint32_t y0 = x0 + k0, y1 = x1 + k1;
#define TFR(r) { y0 += y1; y1 = rotl32(y1, r); y1 ^= y0; }
  TFR(13) TFR(15) TFR(26) TFR(6)   y0 += k1;  y1 += ks2 + 1u;
  TFR(17) TFR(29) TFR(16) TFR(24)  y0 += ks2; y1 += k0  + 2u;
  TFR(13) TFR(15) TFR(26) TFR(6)   y0 += k0;  y1 += k1  + 3u;
  TFR(17) TFR(29) TFR(16) TFR(24)  y0 += k1;  y1 += ks2 + 4u;
  TFR(13) TFR(15) TFR(26) TFR(6)   y0 += ks2; y1 += k0  + 5u;
#undef TFR
  o0 = y0; o1 = y1;
}

__device__ __forceinline__ float bits_to_unit(uint32_t b) {
  // JAX: ((bits >> 9) | 0x3f800000) in [1,2) minus 1 -> [0,1)
  return __uint_as_float((b >> 9) | 0x3f800000u) - 1.0f;
}

__device__ __forceinline__ float gumbel_from_bits(uint32_t b) {
  float u = fmaxf(bits_to_unit(b), 1.1754944e-38f);  // minval = tiny
  return -__logf(-__logf(u));
}

// ---------------- kernels ----------------
__global__ void init_th_kernel(int* th) {
  if (threadIdx.x == 0) *th = 0x7fffffff;
}

// Row sums of Adj -> Dws; first zero-degree row -> th (atomicMin).
__global__ void degree_kernel(const float* __restrict__ Adj,
                              float* __restrict__ Dws,
                              int* __restrict__ th) {
  __shared__ float red[256];
  const int row = blockIdx.x;
  const int tid = threadIdx.x;
  const float* rp = Adj + (size_t)row * N_NODES;
  float s = 0.f;
#pragma unroll
  for (int k = 0; k < N_NODES / 256; ++k) s += rp[tid + 256 * k];
  red[tid] = s;
  __syncthreads();
  for (int o = 128; o > 0; o >>= 1) {
    if (tid < o) red[tid] += red[tid + o];
    __syncthreads();
  }
  if (tid == 0) {
    const float d = red[0];
    Dws[row] = d;
    if (d == 0.f) atomicMin(th, row);
  }
}

// pos[e] = dot(H[i-1], H[j-1]) via V_WMMA_F32_16X16X4_F32 on 16-edge tiles
// (16 dots = diagonal of A·B^T); pi[e] = di/(di+dj). One wave per 16 edges.
// Two b128 gathers per lane per WMMA step (A and B); lane half selects its K-pair.
__global__ void pos_pi_kernel(const int* __restrict__ snap,
                              const float* __restrict__ H,
                              const float* __restrict__ Dws,
                              float* __restrict__ pos,
                              float* __restrict__ pi) {
  __shared__ float tile[16][16];
  const int lane = threadIdx.x;        // 32 threads, EXEC all-ones for WMMA
  const int em   = lane & 15;          // edge-in-tile = M = N for diagonal use
  const int eb   = blockIdx.x * 16;
  const int e    = eb + em;
  const int ia   = snap[2 * e] - 1;
  const int jb   = snap[2 * e + 1] - 1;
  const bool hi  = (lane >= 16);
  const float* Ha = H + (size_t)ia * D_FEAT;
  const float* Hb = H + (size_t)jb * D_FEAT;

  v8f c = {0.f, 0.f, 0.f, 0.f, 0.f, 0.f, 0.f, 0.f};
#pragma unroll 4
  for (int k0 = 0; k0 < D_FEAT; k0 += 4) {
    const float4 av = *(const float4*)(Ha + k0);
    const float4 bv = *(const float4*)(Hb + k0);
    v2f a, b;
    a.x = hi ? av.z : av.x;  a.y = hi ? av.w : av.y;  // A: 16x4, M=lane&15
    b.x = hi ? bv.z : bv.x;  b.y = hi ? bv.w : bv.y;  // B: 4x16, N=lane&15
    c = __builtin_amdgcn_wmma_f32_16x16x4_f32(false, a, false, b,
                                              (short)0, c, false, false);
  }
  const int mb = hi ? 8 : 0;
#pragma unroll
  for (int v = 0; v < 8; ++v) tile[mb + v][em] = c[v];
  __syncthreads();
  if (lane < 16) {
    pos[e] = tile[lane][lane];
    const float di = Dws[ia], dj = Dws[jb];
    pi[e] = di / (di + dj);
  }
}

// One wave per EDGE PAIR (e, e+8192): one threefry per column yields the
// gumbel bits for BOTH edges (JAX pairs counters idx and idx+2^25), halving
// the dominant RNG cost. Per-edge early exit via wave-uniform flags.
__global__ void __launch_bounds__(256)
round_kernel(const float* __restrict__ adj, const int* __restrict__ snap,
             const float* __restrict__ H, const float* __restrict__ pos,
             const float* __restrict__ pi, const int* __restrict__ thp,
             float* __restrict__ out) {
  const int lane = threadIdx.x & 31;
  const int wave = threadIdx.x >> 5;
  const int e0   = blockIdx.x * 8 + wave;   // e0 in [0, 8192)
  const int e1   = e0 + HALF_E;

  // keys = split(key(42), 4): threefry(0,42, iota(8)) -> reshape(4,2)
  uint32_t o0, o1;
  uint32_t kr0[4], kr1[4];
  threefry2x32(0u, 42u, 0u, 4u, o0, o1); kr0[0] = o0; kr0[2] = o1;
  threefry2x32(0u, 42u, 1u, 5u, o0, o1); kr1[0] = o0; kr1[2] = o1;
  threefry2x32(0u, 42u, 2u, 6u, o0, o1); kr0[1] = o0; kr0[3] = o1;
  threefry2x32(0u, 42u, 3u, 7u, o0, o1); kr1[1] = o0; kr1[3] = o1;

  const int thraw = *thp;
  const int thm1  = (thraw == 0x7fffffff) ? 0 : thraw;  // node_id<th <=> col<thm1
  const int tmax  = (thm1 + 127) >> 7;                  // skip cols >= th

  const int i0 = snap[2 * e0], j0 = snap[2 * e0 + 1];
  const int i1 = snap[2 * e1], j1 = snap[2 * e1 + 1];
  const float pi0 = pi[e0], pos0 = pos[e0];
  const float pi1 = pi[e1], pos1 = pos[e1];

  // Warm all four candidate adj rows into cache while we churn RNG keys.
#pragma unroll
  for (int t = 0; t < 4; ++t) {
    __builtin_prefetch(adj + (size_t)(i0 - 1) * N_NODES + t * 1024 + lane * 32, 0, 3);
    __builtin_prefetch(adj + (size_t)(j0 - 1) * N_NODES + t * 1024 + lane * 32, 0, 3);
    __builtin_prefetch(adj + (size_t)(i1 - 1) * N_NODES + t * 1024 + lane * 32, 0, 3);
    __builtin_prefetch(adj + (size_t)(j1 - 1) * N_NODES + t * 1024 + lane * 32, 0, 3);
  }

  const uint32_t base = (uint32_t)e0 * (uint32_t)N_NODES;  // < 2^25 always
  const uint32_t half = ((uint32_t)E_EDGES * (uint32_t)N_NODES) >> 1;  // 2^25

  float res0 = 0.f, res1 = 0.f;
  bool done0 = false, done1 = false;
  for (int r = 0; r < MAX_ROUNDS && !(done0 && done1); ++r) {
    // k1,k2 = split(keys[r], 2)
    uint32_t k1a, k1b, k2a, k2b;
    threefry2x32(kr0[r], kr1[r], 0u, 2u, k1a, k2a);
    threefry2x32(kr0[r], kr1[r], 1u, 3u, k1b, k2b);

    // u for BOTH edges from one call: counters (e0, e0+8192)
    uint32_t ub0, ub1;
    threefry2x32(k1a, k1b, (uint32_t)e0, (uint32_t)e1, ub0, ub1);
    const int d0 = (bits_to_unit(ub0) < pi0) ? j0 : i0;
    const int d1 = (bits_to_unit(ub1) < pi1) ? j1 : i1;
    const float* drow0 = adj + (size_t)(d0 - 1) * N_NODES;
    const float* drow1 = adj + (size_t)(d1 - 1) * N_NODES;

    float best0 = -__builtin_inff(), best1 = -__builtin_inff();
    int bcol0 = 0, bcol1 = 0;
    for (int t = 0; t < tmax; ++t) {
      const int c0 = t * 128 + lane * 4;
      float4 av0 = {0.f, 0.f, 0.f, 0.f}, av1 = {0.f, 0.f, 0.f, 0.f};
      if (!done0) av0 = *(const float4*)(drow0 + c0);
      if (!done1) av1 = *(const float4*)(drow1 + c0);
      const float a00 = av0.x, a01 = av0.y, a02 = av0.z, a03 = av0.w;
      const float a10 = av1.x, a11 = av1.y, a12 = av1.z, a13 = av1.w;
#pragma unroll
      for (int s = 0; s < 4; ++s) {
        const int col = c0 + s;
        const uint32_t idx = base + (uint32_t)col;       // < half, always
        uint32_t gb0, gb1;
        threefry2x32(k2a, k2b, idx, idx + half, gb0, gb1);  // both edges at once
        const float a0 = (s == 0) ? a00 : (s == 1) ? a01 : (s == 2) ? a02 : a03;
        const float a1 = (s == 0) ? a10 : (s == 1) ? a11 : (s == 2) ? a12 : a13;
        if (!done0 && a0 == 0.f && col < thm1 && col != d0 - 1) {
          const float g = gumbel_from_bits(gb0);
          if (g > best0) { best0 = g; bcol0 = col; }
        }
        if (!done1 && a1 == 0.f && col < thm1 && col != d1 - 1) {
          const float g = gumbel_from_bits(gb1);
          if (g > best1) { best1 = g; bcol1 = col; }
        }
      }
    }
    // wave32 butterfly argmax, lexicographic (g desc, col asc)
    for (int off = 16; off > 0; off >>= 1) {
      const float og0 = __shfl_xor(best0, off, 32);
      const int   ob0 = __shfl_xor(bcol0, off, 32);
      if (og0 > best0 || (og0 == best0 && ob0 < bcol0)) { best0 = og0; bcol0 = ob0; }
      const float og1 = __shfl_xor(best1, off, 32);
      const int   ob1 = __shfl_xor(bcol1, off, 32);
      if (og1 > best1 || (og1 == best1 && ob1 < bcol1)) { best1 = og1; bcol1 = ob1; }
    }

    if (!done0) {
      const int dn = bcol0 + 1;
      const int a = min(d0, dn), b = max(d0, dn);
      const float* Pa = H + (size_t)(a - 1) * D_FEAT;
      const float* Pb = H + (size_t)(b - 1) * D_FEAT;
      float s = 0.f;
#pragma unroll
      for (int q = 0; q < D_FEAT / 32; ++q)
        s = fmaf(Pa[lane + 32 * q], Pb[lane + 32 * q], s);
      for (int off = 16; off > 0; off >>= 1) s += __shfl_xor(s, off, 32);
      const float loss = pos0 - s;
      if (loss <= 0.f) { res0 = loss; done0 = true; }
    }
    if (!done1) {
      const int dn = bcol1 + 1;
      const int a = min(d1, dn), b = max(d1, dn);
      const float* Pa = H + (size_t)(a - 1) * D_FEAT;
      const float* Pb = H + (size_t)(b - 1) * D_FEAT;
      float s = 0.f;
#pragma unroll
      for (int q = 0; q < D_FEAT / 32; ++q)
        s = fmaf(Pa[lane + 32 * q], Pb[lane + 32 * q], s);
      for (int off = 16; off > 0; off >>= 1) s += __shfl_xor(s, off, 32);
      const float loss = pos1 - s;
      if (loss <= 0.f) { res1 = loss; done1 = true; }
    }
  }
  if (lane == 0) {
    out[e0] = res0;
    out[e1] = res1;
  }
}

// ---------------- launcher ----------------
extern "C" void kernel_launch(void* const* d_in, const int* in_sizes, int n_in,
                              void* d_out, int out_size, void* d_ws, size_t ws_size,
                              hipStream_t stream) {
  (void)in_sizes; (void)n_in; (void)out_size; (void)ws_size;
  const float* adj  = (const float*)d_in[0];
  const float* Adj  = (const float*)d_in[1];
  const int*   snap = (const int*)d_in[2];
  const float* H    = (const float*)d_in[3];
  float* ws  = (float*)d_ws;
  float* Dws = ws;                          // N_NODES floats
  float* pos = ws + N_NODES;                // E_EDGES floats
  float* pi  = ws + N_NODES + E_EDGES;      // E_EDGES floats
  int*   th  = (int*)(ws + N_NODES + 2 * E_EDGES);
  float* out = (float*)d_out;

  hipLaunchKernelGGL(init_th_kernel, dim3(1), dim3(1), 0, stream, th);
  hipLaunchKernelGGL(degree_kernel, dim3(N_NODES), dim3(256), 0, stream, Adj, Dws, th);
  hipLaunchKernelGGL(pos_pi_kernel, dim3(E_EDGES / 16), dim3(32), 0, stream, snap, H, Dws, pos, pi);
  hipLaunchKernelGGL(round_kernel, dim3(HALF_E / 8), dim3(256), 0, stream,
                     adj, snap, H, pos, pi, th, out);
}